// ColorHead_78099685310947
// MI455X (gfx1250) — compile-verified
//
#include <hip/hip_runtime.h>

// ---------------------------------------------------------------------------
// Problem constants (fixed by the reference: B=2,C=3,H=W=512,F=16, 5 loop iters)
// ---------------------------------------------------------------------------
#define H_  512
#define W_  512
#define HW_ (H_ * W_)          // 262144 = 2^18
#define B_  2
#define C_  3
#define F_  16
#define NBLK 128               // partial-reduction blocks per channel
#define EPSBN 1e-5f
#define NITER 5                // floor(log_3(512)) = 5
#define PW_ 522                // reflect-padded width/height (512 + 2*5)
#define PPLANE (PW_ * PW_)     // 272484

// ws layout (in floats):
//  [0..160)      : per-channel affine params a1,b1,a2,b2,ad,bd,aW,bW,aH,bH (16 each)
//  [256..2304)   : partial sums   pSum[f][NBLK]
//  [2304..4352)  : partial sumsq  pSq [f][NBLK]
//  [8192..417792): reflect-padded u8 level image  (B*C*522*522 = 1,634,904 bytes)
//  [417792..)    : xq   (B*C*HW  = 1,572,864 f)
//  then          : bufA (B*F*HW  = 8,388,608 f)
//  then          : bufB (B*F*HW  = 8,388,608 f)
//  total ~ 18.77M floats = 75.1 MB  (must be <= ws_size)
#define WS_PSUM 256
#define WS_PSQ  2304
#define WS_QPAD 8192
#define WS_XQ   417792

typedef __attribute__((ext_vector_type(2))) float v2f;
typedef __attribute__((ext_vector_type(8))) float v8f;

__device__ __forceinline__ float lrelu(float x) { return x > 0.0f ? x : 0.01f * x; }
__device__ __forceinline__ int reflect512(int p) {
  return p < 0 ? -p : (p > 511 ? 1022 - p : p);
}

// ---------------------------------------------------------------------------
// 0) Quantize once into a reflect-padded u8 level image (L2-resident, 1.6 MB).
//    level = round(x*255/16) in 0..16 ; RNE rintf matches jnp.round.
// ---------------------------------------------------------------------------
__global__ void k_quant(const float* __restrict__ x, unsigned char* __restrict__ qp) {
  int total = B_ * C_ * PPLANE;
  int stride = gridDim.x * blockDim.x;
  for (int i = blockIdx.x * blockDim.x + threadIdx.x; i < total; i += stride) {
    int bc = i / PPLANE;
    int r  = i - bc * PPLANE;
    int py = r / PW_;
    int px = r - py * PW_;
    int y  = reflect512(py - 5);
    int xx = reflect512(px - 5);
    float v = x[(size_t)bc * HW_ + y * W_ + xx];
    qp[i] = (unsigned char)(int)rintf(v * 15.9375f);
  }
}

// ---------------------------------------------------------------------------
// 1) 11x11 sliding-window mode (ties -> smallest level).
//    One thread per (image, column, 64-row segment); per-thread 17-bin LDS
//    histogram (stride-17 -> bank-conflict-free), vertical slide:
//    22 u8 loads + 17-scan per output pixel.
// ---------------------------------------------------------------------------
__global__ void k_mode(const unsigned char* __restrict__ qp, float* __restrict__ xq) {
  __shared__ unsigned int hist[256 * 17];
  int tid = threadIdx.x;
  unsigned int* h = &hist[tid * 17];
#pragma unroll
  for (int l = 0; l < 17; ++l) h[l] = 0;

  int g   = blockIdx.x * blockDim.x + tid;   // 0 .. B*C*512*8-1
  int col = g & 511;
  int seg = (g >> 9) & 7;
  int bc  = g >> 12;                         // 0..5  (b*C + c)
  const unsigned char* q = qp + (size_t)bc * PPLANE;
  float* dst = xq + (size_t)bc * HW_;
  int y0 = seg * 64;

  // window for output y covers padded rows y..y+10, padded cols col..col+10
  for (int dy = 0; dy <= 10; ++dy) {
    const unsigned char* row = q + (y0 + dy) * PW_ + col;
    for (int dx = 0; dx <= 10; ++dx) h[row[dx]]++;
  }
  for (int y = y0; y < y0 + 64; ++y) {
    unsigned int best = 0; int bl = 0;
#pragma unroll
    for (int l = 0; l < 17; ++l) {
      unsigned int c = h[l];
      if (c > best) { best = c; bl = l; }    // strict > keeps smallest level on tie
    }
    dst[y * W_ + col] = (float)bl * 0.0625f; // level * 16/256
    if (y < y0 + 63) {
      const unsigned char* rm = q + y * PW_ + col;          // leaving row
      const unsigned char* ad = q + (y + 11) * PW_ + col;   // entering row
      for (int dx = 0; dx <= 10; ++dx) { h[rm[dx]]--; h[ad[dx]]++; }
    }
  }
}

// ---------------------------------------------------------------------------
// 2) 1x1 conv as WMMA f32 16x16x4:  D(16ch x 16px) = sum_kc A(16x4) B(4x16) + C
//    A layout per ISA table (VGPR0: K=0 lanes0-15 / K=2 lanes16-31; VGPR1: K=1/K=3)
//    B assumed symmetric (VGPR0: K=0/K=2, VGPR1: K=1/K=3, N = lane%16).
//    K positions >= CIN: A carries the hard zero (cndmask); B loads a *clamped*
//    in-bounds channel unconditionally (finite garbage * 0 = 0) -> no EXEC
//    divergence in the tile loop, WMMA sees EXEC all-1s.
// ---------------------------------------------------------------------------
#if __has_builtin(__builtin_amdgcn_wmma_f32_16x16x4_f32)
#define HAVE_WMMA_F32 1
#endif

template <int CIN, int NCH>
__global__ void k_conv1x1_wmma(const float* __restrict__ src, float* __restrict__ dst,
                               const float* __restrict__ w, const float* __restrict__ bias) {
  int lane = threadIdx.x & 31;
  int m    = lane & 15;          // out-channel (A) / pixel (B,D) within tile
  int hi   = lane >> 4;          // half-wave selects K {0,1} vs {2,3} in each chunk
  v2f a[NCH];
  int kb0[NCH], kb1[NCH];
#pragma unroll
  for (int kc = 0; kc < NCH; ++kc) {
    int k0 = 4 * kc + (hi ? 2 : 0);
    int c0 = (k0     < CIN) ? k0     : CIN - 1;   // clamped, always in-bounds
    int c1 = (k0 + 1 < CIN) ? k0 + 1 : CIN - 1;
    float wx = w[m * CIN + c0];
    float wy = w[m * CIN + c1];
    a[kc].x = (k0     < CIN) ? wx : 0.0f;         // true zeros live in A
    a[kc].y = (k0 + 1 < CIN) ? wy : 0.0f;
    kb0[kc] = c0; kb1[kc] = c1;
  }
  int wave   = (blockIdx.x * blockDim.x + threadIdx.x) >> 5;
  int nWaves = (gridDim.x * blockDim.x) >> 5;
  int nTiles = B_ * HW_ / 16;

  for (int t = wave; t < nTiles; t += nWaves) {   // uniform per-wave control flow
    int b  = (t * 16) >> 18;
    int pb = (t * 16) & (HW_ - 1);
    v8f c = {0.f, 0.f, 0.f, 0.f, 0.f, 0.f, 0.f, 0.f};
#pragma unroll
    for (int kc = 0; kc < NCH; ++kc) {
      v2f bv;
      bv.x = src[((size_t)(b * CIN + kb0[kc])) * HW_ + pb + m];
      bv.y = src[((size_t)(b * CIN + kb1[kc])) * HW_ + pb + m];
#if HAVE_WMMA_F32
      c = __builtin_amdgcn_wmma_f32_16x16x4_f32(false, a[kc], false, bv,
                                                (short)0, c, false, false);
#else
      for (int r = 0; r < 8; ++r) c[r] += a[kc].x * bv.x + a[kc].y * bv.y;
#endif
    }
#pragma unroll
    for (int r = 0; r < 8; ++r) {
      int ch = r + hi * 8;                        // D: VGPR r -> M = r (+8)
      dst[((size_t)(b * F_ + ch)) * HW_ + pb + m] = c[r] + bias[ch];
    }
  }
}

// ---------------------------------------------------------------------------
// 3) Deterministic per-channel stats: pass 1 -> per-block partial sum/sumsq.
//    Each block covers 4096 contiguous elements (1024 float4) of one plane.
// ---------------------------------------------------------------------------
__global__ void k_stats(const float* __restrict__ buf, float* __restrict__ ws) {
  __shared__ float s1[256], s2[256];
  int f = blockIdx.y;
  int t = threadIdx.x;
  int bx = blockIdx.x;                           // 0..127 ; 64 blocks per plane
  int b   = bx >> 6;
  int off = (bx & 63) * 4096;
  const float4* p = (const float4*)(buf + (((size_t)(b * F_ + f)) << 18) + off);
  float sum = 0.0f, sq = 0.0f;
  for (int i = t; i < 1024; i += 256) {
    float4 v = p[i];
    sum += v.x + v.y + v.z + v.w;
    sq  += v.x * v.x + v.y * v.y + v.z * v.z + v.w * v.w;
  }
  s1[t] = sum; s2[t] = sq; __syncthreads();
  for (int s = 128; s > 0; s >>= 1) {
    if (t < s) { s1[t] += s1[t + s]; s2[t] += s2[t + s]; }
    __syncthreads();
  }
  if (t == 0) {
    ws[WS_PSUM + f * NBLK + bx] = s1[0];
    ws[WS_PSQ  + f * NBLK + bx] = s2[0];
  }
}

// pass 2: ordered serial sum of partials -> BN affine  out = a*x + b
__global__ void k_finalize_bn(float* ws, const float* __restrict__ g,
                              const float* __restrict__ beta, int aOff, int bOff) {
  int f = threadIdx.x;
  if (f >= F_) return;
  float sum = 0.0f, sq = 0.0f;
  for (int i = 0; i < NBLK; ++i) {
    sum += ws[WS_PSUM + f * NBLK + i];
    sq  += ws[WS_PSQ  + f * NBLK + i];
  }
  const float invN = 1.0f / (float)(B_ * HW_);
  float mean = sum * invN;
  float var  = sq * invN - mean * mean;           // biased var (jnp default)
  float a = g[f] / sqrtf(var + EPSBN);
  ws[aOff + f] = a;
  ws[bOff + f] = beta[f] - mean * a;
}

// pass 2 for width/height branches: stats of (d*s) derive from stats of d:
// mean(d*s)=s*mu, var(d*s)=s^2*var  ->  wdt = lrelu(aW*d + bW)
__global__ void k_finalize_wh(float* ws,
                              const float* __restrict__ ww, const float* __restrict__ gw,
                              const float* __restrict__ bw, const float* __restrict__ wh,
                              const float* __restrict__ gh, const float* __restrict__ bh) {
  int f = threadIdx.x;
  if (f >= F_) return;
  float sum = 0.0f, sq = 0.0f;
  for (int i = 0; i < NBLK; ++i) {
    sum += ws[WS_PSUM + f * NBLK + i];
    sq  += ws[WS_PSQ  + f * NBLK + i];
  }
  const float invN = 1.0f / (float)(B_ * HW_);
  float mean = sum * invN;
  float var  = sq * invN - mean * mean;
  float sW = ww[f];
  float aW = sW * gw[f] / sqrtf(sW * sW * var + EPSBN);
  ws[96  + f] = aW;
  ws[112 + f] = bw[f] - mean * aW;
  float sH = wh[f];
  float aH = sH * gh[f] / sqrtf(sH * sH * var + EPSBN);
  ws[128 + f] = aH;
  ws[144 + f] = bh[f] - mean * aH;
}

// ---------------------------------------------------------------------------
// Elementwise kernels (float4 / b128 memory path; HW is a multiple of 4 so a
// float4 never crosses a channel boundary: ch = (i4 >> 16) & 15)
// ---------------------------------------------------------------------------
__global__ void k_bnlrelu(const float4* __restrict__ src, float4* __restrict__ dst,
                          const float* __restrict__ a, const float* __restrict__ b) {
  size_t total = (size_t)B_ * F_ * HW_ / 4;
  size_t stride = (size_t)gridDim.x * blockDim.x;
  for (size_t i = (size_t)blockIdx.x * blockDim.x + threadIdx.x; i < total; i += stride) {
    int ch = (int)((i >> 16) & 15);
    float aa = a[ch], bb = b[ch];
    float4 v = src[i];
    v.x = lrelu(aa * v.x + bb); v.y = lrelu(aa * v.y + bb);
    v.z = lrelu(aa * v.z + bb); v.w = lrelu(aa * v.w + bb);
    dst[i] = v;
  }
}

// p = bn2+lrelu(t2); keep p as d-state, init inter: inter[ob, b*16+f] = p[b,f], ob=0,1
__global__ void k_p_inter(const float4* __restrict__ t2, float4* __restrict__ d,
                          const float* __restrict__ a, const float* __restrict__ b,
                          float4* __restrict__ out) {
  size_t total = (size_t)B_ * F_ * HW_ / 4;
  size_t stride = (size_t)gridDim.x * blockDim.x;
  for (size_t i = (size_t)blockIdx.x * blockDim.x + threadIdx.x; i < total; i += stride) {
    int bf = (int)(i >> 16);
    int ch = bf & 15, bat = bf >> 4;
    float aa = a[ch], bb = b[ch];
    float4 v = t2[i];
    v.x = lrelu(aa * v.x + bb); v.y = lrelu(aa * v.y + bb);
    v.z = lrelu(aa * v.z + bb); v.w = lrelu(aa * v.w + bb);
    d[i] = v;
    size_t pix4 = i & 0xFFFF;                          // HW/4 - 1
    size_t o = (((size_t)(bat * F_ + ch)) << 16) + pix4;
    out[o] = v;                                        // out batch 0
    out[o + ((size_t)32 << 16)] = v;                   // out batch 1 (identical init)
  }
}

// depthwise 3x3, zero ('SAME') padding, cross-correlation; 4 outputs per thread
__global__ void k_dwconv(const float* __restrict__ d, float4* __restrict__ t,
                         const float* __restrict__ dwv) {
  size_t total = (size_t)B_ * F_ * HW_ / 4;
  size_t stride = (size_t)gridDim.x * blockDim.x;
  for (size_t i = (size_t)blockIdx.x * blockDim.x + threadIdx.x; i < total; i += stride) {
    int bf = (int)(i >> 16);
    int f  = bf & 15;
    int p4 = (int)(i & 0xFFFF);
    int y  = p4 >> 7;                 // 128 float4 per row
    int x0 = (p4 & 127) * 4;
    const float* plane = d + ((size_t)bf << 18);
    float wk[9];
#pragma unroll
    for (int k = 0; k < 9; ++k) wk[k] = dwv[f * 9 + k];
    float4 acc = make_float4(0.f, 0.f, 0.f, 0.f);
#pragma unroll
    for (int ky = 0; ky < 3; ++ky) {
      int yy = y + ky - 1;
      if (yy < 0 || yy > 511) continue;               // zero pad rows
      const float* row = plane + yy * W_;
      float4 c = *(const float4*)(row + x0);
      float l = (x0 > 0)   ? row[x0 - 1] : 0.0f;      // zero pad cols
      float r = (x0 < 508) ? row[x0 + 4] : 0.0f;
      float w0 = wk[ky * 3], w1 = wk[ky * 3 + 1], w2 = wk[ky * 3 + 2];
      acc.x += w0 * l   + w1 * c.x + w2 * c.y;
      acc.y += w0 * c.x + w1 * c.y + w2 * c.z;
      acc.z += w0 * c.y + w1 * c.z + w2 * c.w;
      acc.w += w0 * c.z + w1 * c.w + w2 * r;
    }
    t[i] = acc;
  }
}

// inter[b, ch] += lrelu(aW*d+bW);  inter[b, 16+ch] += lrelu(aH*d+bH)
__global__ void k_accum(const float4* __restrict__ d, float4* __restrict__ out,
                        const float* __restrict__ aW, const float* __restrict__ bW,
                        const float* __restrict__ aH, const float* __restrict__ bH) {
  size_t total = (size_t)B_ * F_ * HW_ / 4;
  size_t stride = (size_t)gridDim.x * blockDim.x;
  for (size_t i = (size_t)blockIdx.x * blockDim.x + threadIdx.x; i < total; i += stride) {
    int bf = (int)(i >> 16);
    int ch = bf & 15, bat = bf >> 4;
    size_t pix4 = i & 0xFFFF;
    float4 dv = d[i];
    size_t o0 = (((size_t)(bat * 32 + ch)) << 16) + pix4;
    size_t o1 = o0 + ((size_t)16 << 16);
    float a0 = aW[ch], b0 = bW[ch], a1 = aH[ch], b1 = bH[ch];
    float4 u = out[o0];
    u.x += lrelu(a0 * dv.x + b0); u.y += lrelu(a0 * dv.y + b0);
    u.z += lrelu(a0 * dv.z + b0); u.w += lrelu(a0 * dv.w + b0);
    out[o0] = u;
    float4 v = out[o1];
    v.x += lrelu(a1 * dv.x + b1); v.y += lrelu(a1 * dv.y + b1);
    v.z += lrelu(a1 * dv.z + b1); v.w += lrelu(a1 * dv.w + b1);
    out[o1] = v;
  }
}

// ---------------------------------------------------------------------------
// Orchestration
// ---------------------------------------------------------------------------
extern "C" void kernel_launch(void* const* d_in, const int* in_sizes, int n_in,
                              void* d_out, int out_size, void* d_ws, size_t ws_size,
                              hipStream_t stream) {
  const float* x   = (const float*)d_in[0];
  const float* w1  = (const float*)d_in[1];
  const float* b1  = (const float*)d_in[2];
  const float* g1  = (const float*)d_in[3];
  const float* be1 = (const float*)d_in[4];
  const float* w2  = (const float*)d_in[5];
  const float* b2  = (const float*)d_in[6];
  const float* g2  = (const float*)d_in[7];
  const float* be2 = (const float*)d_in[8];
  const float* dw  = (const float*)d_in[9];
  const float* gd  = (const float*)d_in[10];
  const float* bd  = (const float*)d_in[11];
  const float* ww  = (const float*)d_in[12];
  const float* gw  = (const float*)d_in[13];
  const float* bw  = (const float*)d_in[14];
  const float* wh  = (const float*)d_in[15];
  const float* gh  = (const float*)d_in[16];
  const float* bh  = (const float*)d_in[17];

  float* ws   = (float*)d_ws;
  float* out  = (float*)d_out;
  unsigned char* qpad = (unsigned char*)(ws + WS_QPAD);
  float* xq   = ws + WS_XQ;
  float* bufA = xq + (size_t)B_ * C_ * HW_;
  float* bufB = bufA + (size_t)B_ * F_ * HW_;

  dim3 sgrid(NBLK, F_);

  // quantize (reflect-padded u8 levels) + mode pool
  k_quant<<<1600, 256, 0, stream>>>(x, qpad);
  k_mode<<<(B_ * C_ * 512 * 8) / 256, 256, 0, stream>>>(qpad, xq);
  // conv1 (WMMA f32 16x16x4, K=3 zero-padded to 4)
  k_conv1x1_wmma<3, 1><<<512, 256, 0, stream>>>(xq, bufA, w1, b1);
  k_stats<<<sgrid, 256, 0, stream>>>(bufA, ws);
  k_finalize_bn<<<1, 32, 0, stream>>>(ws, g1, be1, 0, 16);
  k_bnlrelu<<<2048, 256, 0, stream>>>((const float4*)bufA, (float4*)bufB, ws + 0, ws + 16);
  // conv2 (WMMA, K=16 as 4 chunks)
  k_conv1x1_wmma<16, 4><<<512, 256, 0, stream>>>(bufB, bufA, w2, b2);
  k_stats<<<sgrid, 256, 0, stream>>>(bufA, ws);
  k_finalize_bn<<<1, 32, 0, stream>>>(ws, g2, be2, 32, 48);
  k_p_inter<<<2048, 256, 0, stream>>>((const float4*)bufA, (float4*)bufB,
                                      ws + 32, ws + 48, (float4*)out);

  for (int it = 0; it < NITER; ++it) {
    k_dwconv<<<2048, 256, 0, stream>>>(bufB, (float4*)bufA, dw);        // t
    k_stats<<<sgrid, 256, 0, stream>>>(bufA, ws);
    k_finalize_bn<<<1, 32, 0, stream>>>(ws, gd, bd, 64, 80);
    k_bnlrelu<<<2048, 256, 0, stream>>>((const float4*)bufA, (float4*)bufB,
                                        ws + 64, ws + 80);              // d
    k_stats<<<sgrid, 256, 0, stream>>>(bufB, ws);
    k_finalize_wh<<<1, 32, 0, stream>>>(ws, ww, gw, bw, wh, gh, bh);
    k_accum<<<2048, 256, 0, stream>>>((const float4*)bufB, (float4*)out,
                                      ws + 96, ws + 112, ws + 128, ws + 144);
  }
}